// SocialLSTMClassifier_14370960572579
// MI455X (gfx1250) — compile-verified
//
#include <hip/hip_runtime.h>
#include <math.h>

#define H   128
#define F   4
#define T   16
#define NN  131072
#define GX_ 4
#define GY_ 4
#define NS_ 4.0f

typedef float v2f __attribute__((ext_vector_type(2)));
typedef float v8f __attribute__((ext_vector_type(8)));

// ---- fast transcendentals: native V_TANH_F32 when available ---------------
__device__ __forceinline__ float fast_tanh(float x) {
#if __has_builtin(__builtin_amdgcn_tanhf)
  return __builtin_amdgcn_tanhf(x);
#else
  float e = __expf(2.0f * x);
  return 1.0f - 2.0f * __builtin_amdgcn_rcpf(e + 1.0f);
#endif
}
__device__ __forceinline__ float fast_sig(float x) {
#if __has_builtin(__builtin_amdgcn_tanhf)
  return fmaf(0.5f, __builtin_amdgcn_tanhf(0.5f * x), 0.5f);   // 1 trans op
#else
  return __builtin_amdgcn_rcpf(1.0f + __expf(-x));
#endif
}

// ---------------------------------------------------------------------------
// Kernel 1: sequential target LSTM (T=16 steps) + zero the global grid accum.
// ws layout (floats): [0..2047] grid accum, [2048..2175] h_T
// ---------------------------------------------------------------------------
__global__ void k_target_lstm(const float* __restrict__ traj,   // T x F
                              const float* __restrict__ W_ih,   // 4H x F
                              const float* __restrict__ W_hh,   // 4H x H
                              const float* __restrict__ b_ih,   // 4H
                              const float* __restrict__ b_hh,   // 4H
                              float* __restrict__ ws) {
  __shared__ float h_s[H], c_s[H], g_s[4 * H];
  const int j = threadIdx.x;                 // 0..511
  ws[j] = 0.f; ws[512 + j] = 0.f; ws[1024 + j] = 0.f; ws[1536 + j] = 0.f;
  if (j < H) { h_s[j] = 0.f; c_s[j] = 0.f; }
  __syncthreads();

  const float bias = b_ih[j] + b_hh[j];
  for (int t = 0; t < T; ++t) {
    float acc = bias;
#pragma unroll
    for (int k = 0; k < F; ++k) acc += traj[t * F + k] * W_ih[j * F + k];
    const float* wr = W_hh + (size_t)j * H;
    for (int k = 0; k < H; ++k) acc += h_s[k] * wr[k];
    g_s[j] = acc;
    __syncthreads();
    if (j < H) {
      float ig = fast_sig(g_s[j]);
      float fg = fast_sig(g_s[H + j]);
      float gg = fast_tanh(g_s[2 * H + j]);
      float og = fast_sig(g_s[3 * H + j]);
      float cn = fg * c_s[j] + ig * gg;
      c_s[j] = cn;
      h_s[j] = og * fast_tanh(cn);
    }
    __syncthreads();
  }
  if (j < H) ws[2048 + j] = h_s[j];
}

// ---------------------------------------------------------------------------
// Kernel 2: fused others-LSTM (h=c=0 => only i,g,o gates) + social grid.
// WMMA f32 16x16x4 with a shared zero-C; bias added in the elementwise phase.
// B fragments/biases hoisted; launch_bounds relaxes the VGPR budget (no spill).
// ---------------------------------------------------------------------------
__global__ void __launch_bounds__(256, 1)
k_others_grid(const float* __restrict__ others, // T x N x F
              const int*   __restrict__ nmask,  // T x N
              const float* __restrict__ traj,   // T x F
              const float* __restrict__ W_ih,   // 4H x F
              const float* __restrict__ b_ih,
              const float* __restrict__ b_hh,
              float* __restrict__ ws) {
  __shared__ float gridl[GX_ * GY_ * H];     // 16 cells x 128 = 8 KB
  const int tid = threadIdx.x;
  for (int i = tid; i < GX_ * GY_ * H; i += blockDim.x) gridl[i] = 0.f;
  __syncthreads();

  const float* X = others + (size_t)(T - 1) * NN * F;   // last timestep
  const int*   M = nmask  + (size_t)(T - 1) * NN;
  const float refx = traj[(T - 1) * F + 0];
  const float refy = traj[(T - 1) * F + 1];

  const int lane  = tid & 31;
  const int wid   = tid >> 5;
  const int gwave = blockIdx.x * (blockDim.x >> 5) + wid;
  const int nwaves = gridDim.x * (blockDim.x >> 5);
  const int ntiles = NN / 16;                // 8192: uniform trip count

  const int mrow  = lane & 15;               // row (A) / column (B,D) selector
  const int khalf = lane >> 4;               // K half: features {0,1} or {2,3}

  // --- tile-invariant B fragments and biases in registers ------------------
  v2f   BI[8], BG[8], BO[8];
  float bI[8], bG[8], bO[8];
#pragma unroll
  for (int jg = 0; jg < 8; ++jg) {
    const int nI = 16 * jg, nG = 256 + 16 * jg, nO = 384 + 16 * jg;
    BI[jg] = *(const v2f*)(W_ih + (size_t)(nI + mrow) * F + 2 * khalf);
    BG[jg] = *(const v2f*)(W_ih + (size_t)(nG + mrow) * F + 2 * khalf);
    BO[jg] = *(const v2f*)(W_ih + (size_t)(nO + mrow) * F + 2 * khalf);
    bI[jg] = b_ih[nI + mrow] + b_hh[nI + mrow];
    bG[jg] = b_ih[nG + mrow] + b_hh[nG + mrow];
    bO[jg] = b_ih[nO + mrow] + b_hh[nO + mrow];
  }
  const v8f zc = {};                         // shared zero C operand

  for (int tile = gwave; tile < ntiles; tile += nwaves) {
    const int r0 = tile * 16;
    // A fragment: lane l holds x[r0 + l%16][2*(l/16)], [2*(l/16)+1] (streamed)
    v2f a = __builtin_nontemporal_load(
        (const v2f*)(X + (size_t)(r0 + mrow) * F + 2 * khalf));

    // Per-row cell index; lanes 0..15 hold pos.x/pos.y in a.x/a.y
    int mycell;
    {
      float rx = a.x - refx;
      float ry = a.y - refy;
      int cx = (int)truncf(rx / (NS_ / GX_)) + GX_ / 2;
      int cy = (int)truncf(ry / (NS_ / GY_)) + GY_ / 2;
      int mk = __builtin_nontemporal_load(M + r0 + mrow);
      bool valid = (fabsf(rx) <= NS_ * 0.5f) && (fabsf(ry) <= NS_ * 0.5f) &&
                   (cx >= 0) && (cx < GX_) && (cy >= 0) && (cy < GY_) && (mk != 0);
      mycell = valid ? (cy * GX_ + cx) : -1;
    }
    // Distribute row cells once per tile (independent of jg): 8 bpermutes.
    int cellv[8];
#pragma unroll
    for (int v = 0; v < 8; ++v) cellv[v] = __shfl(mycell, v + 8 * khalf, 32);

#pragma unroll
    for (int jg = 0; jg < 8; ++jg) {         // hidden groups of 16
      v8f aI = __builtin_amdgcn_wmma_f32_16x16x4_f32(false, a, false, BI[jg], (short)0, zc, false, false);
      v8f aG = __builtin_amdgcn_wmma_f32_16x16x4_f32(false, a, false, BG[jg], (short)0, zc, false, false);
      v8f aO = __builtin_amdgcn_wmma_f32_16x16x4_f32(false, a, false, BO[jg], (short)0, zc, false, false);

      // D layout: vgpr v, lane l -> row = v + 8*(l/16), col = n0 + l%16
#pragma unroll
      for (int v = 0; v < 8; ++v) {
        float si = fast_sig(aI[v] + bI[jg]);
        float tg = fast_tanh(aG[v] + bG[jg]);
        float so = fast_sig(aO[v] + bO[jg]);
        float hv = so * fast_tanh(si * tg);
        int   c_ = cellv[v];
        float hvm = (c_ >= 0) ? hv : 0.0f;   // branchless: add 0 if invalid
        int   idx = ((c_ >= 0) ? c_ : 0) * H + 16 * jg + mrow;
        atomicAdd(&gridl[idx], hvm);
      }
    }
  }
  __syncthreads();
  for (int i = tid; i < GX_ * GY_ * H; i += blockDim.x) atomicAdd(&ws[i], gridl[i]);
}

// ---------------------------------------------------------------------------
// Kernel 3: head. social(2048) -> relu(W1) -> W2 -> +h_T -> Wc -> out(2)
// ---------------------------------------------------------------------------
__global__ void k_head(const float* __restrict__ W1, const float* __restrict__ b1,
                       const float* __restrict__ W2, const float* __restrict__ b2,
                       const float* __restrict__ Wc, const float* __restrict__ bc,
                       const float* __restrict__ ws, float* __restrict__ out) {
  __shared__ float r1[H], comb[H];
  const int m = threadIdx.x;                 // 0..127
  float acc = b1[m];
  const float* w = W1 + (size_t)m * (H * GX_ * GY_);
  for (int k = 0; k < H * GX_ * GY_; ++k) acc += ws[k] * w[k];
  r1[m] = fmaxf(acc, 0.f);
  __syncthreads();
  float acc2 = b2[m];
  const float* w2 = W2 + (size_t)m * H;
  for (int k = 0; k < H; ++k) acc2 += r1[k] * w2[k];
  comb[m] = ws[2048 + m] + acc2;
  __syncthreads();
  if (m < 2) {
    float o = bc[m];
    const float* wc = Wc + (size_t)m * H;
    for (int k = 0; k < H; ++k) o += comb[k] * wc[k];
    out[m] = o;
  }
}

extern "C" void kernel_launch(void* const* d_in, const int* in_sizes, int n_in,
                              void* d_out, int out_size, void* d_ws, size_t ws_size,
                              hipStream_t stream) {
  (void)in_sizes; (void)n_in; (void)out_size; (void)ws_size;
  const float* traj  = (const float*)d_in[0];   // T x F
  const float* oth   = (const float*)d_in[1];   // T x N x F
  const int*   msk   = (const int*)  d_in[2];   // T x N
  const float* W_ih  = (const float*)d_in[3];   // 4H x F
  const float* W_hh  = (const float*)d_in[4];   // 4H x H
  const float* b_ih  = (const float*)d_in[5];
  const float* b_hh  = (const float*)d_in[6];
  const float* W1    = (const float*)d_in[7];   // H x (H*16)
  const float* b1    = (const float*)d_in[8];
  const float* W2    = (const float*)d_in[9];   // H x H
  const float* b2    = (const float*)d_in[10];
  const float* Wc    = (const float*)d_in[11];  // 2 x H
  const float* bc    = (const float*)d_in[12];
  float* ws  = (float*)d_ws;
  float* out = (float*)d_out;

  k_target_lstm<<<1, 4 * H, 0, stream>>>(traj, W_ih, W_hh, b_ih, b_hh, ws);
  k_others_grid<<<256, 256, 0, stream>>>(oth, msk, traj, W_ih, b_ih, b_hh, ws);
  k_head<<<1, H, 0, stream>>>(W1, b1, W2, b2, Wc, bc, ws, out);
}